// ChamferDistance_80582176408029
// MI455X (gfx1250) — compile-verified
//
#include <hip/hip_runtime.h>

// Chamfer distance on MI455X (gfx1250, wave32).
// dist(n,m) = |x_n|^2 + |y_m|^2 - 2 * <x_n, y_m>; cross term via
// V_WMMA_F32_16X16X4_F32 (16x4 A * 4x16 B -> 16x16 f32), K padded 3->4
// with A's pad column = 0 (so B's K=3 row is don't-care).
//
// Per workgroup: 256 threads = 8 wave32, each wave owns one 16-row tile of X.
// The opposing set Y is staged through LDS in 1024-point chunks as float4
// (x, y, z, |y|^2) so the squared norm arrives with the same ds_load_b128.

typedef __attribute__((ext_vector_type(2))) float v2f;
typedef __attribute__((ext_vector_type(8))) float v8f;

#define CHUNK 1024          // Y points staged in LDS per iteration (16 KB)
#define BLOCK 256           // 8 waves
#define ROWS_PER_BLOCK 128  // 8 waves * 16 rows

__global__ __launch_bounds__(BLOCK)
void chamfer_min_kernel(const float* __restrict__ X,   // [B, NX, 3]
                        const float* __restrict__ Y,   // [B, NY, 3]
                        int NX, int NY,
                        float* __restrict__ outMin)    // [B, NX]
{
    __shared__ float4 Ylds[CHUNK];

    const int tilesPerBatch = NX / ROWS_PER_BLOCK;
    const int b    = blockIdx.x / tilesPerBatch;
    const int n0   = (blockIdx.x % tilesPerBatch) * ROWS_PER_BLOCK
                   + ((threadIdx.x >> 5) << 4);        // wave's 16-row tile base
    const int lane = threadIdx.x & 31;
    const int hlf  = lane >> 4;                        // half-wave: 0 or 1
    const int lid  = lane & 15;

    const float* __restrict__ Xb = X + (size_t)b * NX * 3;
    const float* __restrict__ Yb = Y + (size_t)b * NY * 3;

    // ---- A-tile (16x4, K-padded). ISA layout: lanes 0-15 hold K=0,1 in
    // VGPR0,1 for M=lane; lanes 16-31 hold K=2,3 for M=lane-16.
    const int arow = n0 + lid;
    v2f a;
    if (hlf == 0) { a.x = Xb[arow * 3 + 0]; a.y = Xb[arow * 3 + 1]; }
    else          { a.x = Xb[arow * 3 + 2]; a.y = 0.0f; }          // pad K=3

    // ---- |x_n|^2 for the 8 C-rows this lane accumulates:
    // C VGPR r: lanes 0-15 -> row r, lanes 16-31 -> row r+8.
    float sqrow[8];
#pragma unroll
    for (int r = 0; r < 8; ++r) {
        const int n = n0 + hlf * 8 + r;
        const float x = Xb[n * 3 + 0], y = Xb[n * 3 + 1], z = Xb[n * 3 + 2];
        sqrow[r] = x * x + y * y + z * z;
    }

    float acc[8];
#pragma unroll
    for (int r = 0; r < 8; ++r) acc[r] = 3.402823466e38f;

    for (int base = 0; base < NY; base += CHUNK) {
        __syncthreads();   // protect Ylds from previous sweep
        // Cooperative stage: 1024 points, 4 per thread, with |y|^2 in .w
        for (int t = threadIdx.x; t < CHUNK; t += BLOCK) {
            const int g = base + t;
            const float x = Yb[g * 3 + 0], y = Yb[g * 3 + 1], z = Yb[g * 3 + 2];
            Ylds[t] = make_float4(x, y, z, x * x + y * y + z * z);
        }
        __syncthreads();

        for (int mt = 0; mt < CHUNK / 16; ++mt) {
            // B-tile (4x16): VGPR j, lanes 0-15 -> K=j; lanes 16-31 -> K=j+2.
            const float4 yv = Ylds[mt * 16 + lid];
            v2f bv;
            bv.x = hlf ? yv.z : yv.x;
            bv.y = hlf ? yv.w : yv.y;    // K=3 row: don't-care (A pad is 0)
            const float sq2l = yv.w;

            v8f c = {};
            c = __builtin_amdgcn_wmma_f32_16x16x4_f32(
                    /*neg_a=*/false, a, /*neg_b=*/false, bv,
                    /*c_mod=*/(short)0, c, /*reuse_a=*/false, /*reuse_b=*/false);

#pragma unroll
            for (int r = 0; r < 8; ++r) {
                float d = fmaf(-2.0f, c[r], sqrow[r] + sq2l);
                d = fmaxf(d, 0.0f);                 // fp-cancellation guard
                acc[r] = fminf(acc[r], d);
            }
        }
    }

    // ---- min across the 16 lanes of this half-wave (columns of the row)
#pragma unroll
    for (int r = 0; r < 8; ++r) {
        float v = acc[r];
        v = fminf(v, __shfl_xor(v, 1, 32));
        v = fminf(v, __shfl_xor(v, 2, 32));
        v = fminf(v, __shfl_xor(v, 4, 32));
        v = fminf(v, __shfl_xor(v, 8, 32));
        acc[r] = v;
    }
    if (lid == 0) {
#pragma unroll
        for (int r = 0; r < 8; ++r)
            outMin[(size_t)b * NX + n0 + hlf * 8 + r] = acc[r];
    }
}

extern "C" void kernel_launch(void* const* d_in, const int* in_sizes, int n_in,
                              void* d_out, int out_size, void* d_ws, size_t ws_size,
                              hipStream_t stream) {
    const float* in1 = (const float*)d_in[0];   // [B, N, 3]
    const float* in2 = (const float*)d_in[1];   // [B, M, 3]
    float* out = (float*)d_out;                 // dist1 [B,N] ++ dist2 [B,M]

    const int B = 8;
    const int N = in_sizes[0] / (B * 3);        // 8192
    const int M = in_sizes[1] / (B * 3);        // 8192

    dim3 block(BLOCK);
    // dist1: for each point of set1, min over set2
    chamfer_min_kernel<<<dim3(B * (N / ROWS_PER_BLOCK)), block, 0, stream>>>(
        in1, in2, N, M, out);
    // dist2: symmetric, swap roles
    chamfer_min_kernel<<<dim3(B * (M / ROWS_PER_BLOCK)), block, 0, stream>>>(
        in2, in1, M, N, out + (size_t)B * N);
}